// GATv2_80796924772445
// MI455X (gfx1250) — compile-verified
//
#include <hip/hip_runtime.h>
#include <hip/hip_bf16.h>

// ---------------------------------------------------------------------------
// GATv2 (B=2, N=192, IN_F=16, NH=8, HID=128, NC=16) for gfx1250 (MI455X).
// fp32 reference -> use V_WMMA_F32_16X16X4_F32 for all dense GEMMs.
// GATv2 score tensor has the nonlinearity inside the reduction -> VALU kernel.
// ---------------------------------------------------------------------------

typedef __attribute__((ext_vector_type(2))) float v2f;
typedef __attribute__((ext_vector_type(8))) float v8f;

namespace cfg {
constexpr int B    = 2;
constexpr int N    = 192;
constexpr int IN_F = 16;
constexpr int NH   = 8;
constexpr int HID  = 128;
constexpr int NC   = 16;
constexpr int M    = B * N;          // 384 rows for all GEMMs
constexpr float SLOPE = 0.2f;
}

// ---------------------------------------------------------------------------
// Generic row-major fp32 GEMM: C[M,N] = act(A[M,K] @ B[K,N] + bias)
// One wave (32 threads) per 16x16 tile; K stepped by 4 via WMMA f32 16x16x4.
// Fragment layouts per CDNA5 ISA 7.12.2:
//   A 16x4 : lane&15 = row M, (lane>>4)*2 = K base; VGPR0/1 = K+0 / K+1
//   B 4x16 : lane&15 = col N, (lane>>4)*2 = K base; VGPR0/1 = K+0 / K+1
//   C 16x16: VGPR r -> row (m0 + (lane>>4)*8 + r), col (n0 + (lane&15))
// ACT: 0 = none, 1 = relu
// ---------------------------------------------------------------------------
template <int ACT>
__global__ void wmma_gemm_f32(const float* __restrict__ A,
                              const float* __restrict__ Bm,
                              const float* __restrict__ bias,
                              float* __restrict__ C,
                              int M, int N, int K) {
    const int lane = threadIdx.x;            // blockDim.x == 32, one wave
    const int n0   = blockIdx.x * 16;
    const int m0   = blockIdx.y * 16;
    const int r    = lane & 15;
    const int kb   = (lane >> 4) << 1;       // 0 or 2

    const float* Ap = A  + (size_t)(m0 + r) * K + kb;  // A[m0+r, kb + k0 + {0,1}]
    const float* Bp = Bm + (size_t)kb * N + (n0 + r);  // B[kb + k0 + {0,1}, n0+r]

    v8f acc = {};
    for (int k0 = 0; k0 < K; k0 += 4) {
        v2f av, bv;
        av.x = Ap[k0];
        av.y = Ap[k0 + 1];
        bv.x = Bp[(size_t)k0 * N];
        bv.y = Bp[(size_t)(k0 + 1) * N];
        acc = __builtin_amdgcn_wmma_f32_16x16x4_f32(
            /*neg_a=*/false, av, /*neg_b=*/false, bv,
            /*c_mod=*/(short)0, acc, /*reuse_a=*/false, /*reuse_b=*/false);
    }

    const float bv = bias ? bias[n0 + r] : 0.0f;
    const int rowbase = m0 + ((lane >> 4) << 3);
    #pragma unroll
    for (int rr = 0; rr < 8; ++rr) {
        float v = acc[rr] + bv;
        if (ACT == 1) v = fmaxf(v, 0.0f);
        C[(size_t)(rowbase + rr) * N + (n0 + r)] = v;
    }
}

// ---------------------------------------------------------------------------
// GATv2 attention for one (b, head, i): scores over j, softmax, aggregate.
// blockDim.x = 128 (= HID). grid = (N, NH, B).
//   e[j]   = sum_f lrelu(g[b,i,h,f] + g[b,j,h,f]) * a[f]
//   e[j]   = -inf where adj==0; = -FLT_MAX where mask_i != mask_j (layer 1)
//   att    = softmax_j(e)
//   agg[b,i,h,f] = sum_j att[j] * g[b,j,h,f]
// ---------------------------------------------------------------------------
__global__ void gat_attention(const float* __restrict__ g,
                              const int*   __restrict__ adj,
                              const int*   __restrict__ mask,
                              const float* __restrict__ a,
                              float* __restrict__ agg,
                              int use_mask) {
    using namespace cfg;
    const int i   = blockIdx.x;
    const int hh  = blockIdx.y;
    const int b   = blockIdx.z;
    const int tid = threadIdx.x;             // 0..127

    __shared__ float sgi[HID];
    __shared__ float se[N];
    __shared__ float red[128];

    const float* gb = g + (size_t)b * N * NH * HID + (size_t)hh * HID;  // + j*NH*HID + f

    sgi[tid] = gb[(size_t)i * NH * HID + tid];
    __syncthreads();

    const int* adji = adj + ((size_t)b * N + i) * N;
    const int  mi   = use_mask ? mask[b * N + i] : 0;
    const float NEG_INF = -__builtin_inff();
    const float FMIN    = -3.40282346638528859812e+38f;  // float32 finfo.min

    for (int j = tid; j < N; j += 128) {
        float e;
        if (adji[j] == 0) {
            e = NEG_INF;
        } else {
            const float* gj = gb + (size_t)j * NH * HID;
            float acc = 0.0f;
            #pragma unroll 4
            for (int f = 0; f < HID; ++f) {
                float s  = sgi[f] + gj[f];
                float ls = (s > 0.0f) ? s : SLOPE * s;
                acc = fmaf(ls, a[f], acc);
            }
            e = acc;
        }
        if (use_mask && (mask[b * N + j] != mi)) e = FMIN;  // applied after adj fill
        se[j] = e;
    }
    __syncthreads();

    // softmax over j: max reduce
    float lm = NEG_INF;
    for (int j = tid; j < N; j += 128) lm = fmaxf(lm, se[j]);
    red[tid] = lm;
    __syncthreads();
    for (int s = 64; s > 0; s >>= 1) {
        if (tid < s) red[tid] = fmaxf(red[tid], red[tid + s]);
        __syncthreads();
    }
    const float maxv = red[0];
    __syncthreads();

    // exp + sum reduce
    float lsum = 0.0f;
    for (int j = tid; j < N; j += 128) {
        float ex = __expf(se[j] - maxv);
        se[j] = ex;
        lsum += ex;
    }
    red[tid] = lsum;
    __syncthreads();
    for (int s = 64; s > 0; s >>= 1) {
        if (tid < s) red[tid] += red[tid + s];
        __syncthreads();
    }
    const float inv = 1.0f / red[0];
    __syncthreads();

    // aggregate: thread tid owns feature f = tid
    const float* gf = gb + tid;
    float o = 0.0f;
    for (int j = 0; j < N; ++j) o = fmaf(se[j], gf[(size_t)j * NH * HID], o);
    agg[(((size_t)b * N + i) * NH + hh) * HID + tid] = o * inv;
}

// ---------------------------------------------------------------------------
// Mean over heads (+ optional ELU): hout[b,n,f] = act(mean_h agg[b,n,h,f])
// ---------------------------------------------------------------------------
__global__ void head_mean_act(const float* __restrict__ agg,
                              float* __restrict__ hout, int apply_elu) {
    using namespace cfg;
    const int idx = blockIdx.x * blockDim.x + threadIdx.x;
    if (idx >= M * HID) return;
    const int f  = idx % HID;
    const int bn = idx / HID;
    float s = 0.0f;
    #pragma unroll
    for (int h = 0; h < NH; ++h) s += agg[((size_t)bn * NH + h) * HID + f];
    s *= (1.0f / NH);
    if (apply_elu) s = (s > 0.0f) ? s : (__expf(s) - 1.0f);
    hout[idx] = s;
}

// ---------------------------------------------------------------------------
// hcat[b,n, :] = [ h3[b,n,0:128], x[b,n,{1,2,6,7}] ]     (HID+4 = 132 cols)
// ---------------------------------------------------------------------------
__global__ void concat_residual(const float* __restrict__ h3,
                                const float* __restrict__ x,
                                float* __restrict__ hcat) {
    using namespace cfg;
    const int idx = blockIdx.x * blockDim.x + threadIdx.x;
    if (idx >= M * (HID + 4)) return;
    const int c  = idx % (HID + 4);
    const int bn = idx / (HID + 4);
    float v;
    if (c < HID) {
        v = h3[(size_t)bn * HID + c];
    } else {
        const int cm[4] = {1, 2, 6, 7};
        v = x[(size_t)bn * IN_F + cm[c - HID]];
    }
    hcat[idx] = v;
}

// ---------------------------------------------------------------------------
extern "C" void kernel_launch(void* const* d_in, const int* in_sizes, int n_in,
                              void* d_out, int out_size, void* d_ws, size_t ws_size,
                              hipStream_t stream) {
    using namespace cfg;
    const float* x    = (const float*)d_in[0];
    const int*   adj  = (const int*)  d_in[1];
    const int*   mask = (const int*)  d_in[2];
    const float* W1   = (const float*)d_in[3];
    const float* a1   = (const float*)d_in[4];
    const float* W2   = (const float*)d_in[5];
    const float* a2   = (const float*)d_in[6];
    const float* Wl1  = (const float*)d_in[7];
    const float* bl1  = (const float*)d_in[8];
    const float* Wl2  = (const float*)d_in[9];
    const float* bl2  = (const float*)d_in[10];
    const float* Wl3  = (const float*)d_in[11];
    const float* bl3  = (const float*)d_in[12];
    float*       out  = (float*)d_out;

    // workspace layout (floats)
    float* ws   = (float*)d_ws;
    float* g    = ws;                               // [B,N,NH,HID] = 393216
    float* agg  = g    + (size_t)M * NH * HID;      // [B,N,NH,HID] = 393216
    float* h    = agg  + (size_t)M * NH * HID;      // [B,N,HID]    = 49152
    float* hcat = h    + (size_t)M * HID;           // [B,N,HID+4]  = 50688
    float* m1   = hcat + (size_t)M * (HID + 4);     // [B,N,HID]    = 49152
    float* m2   = m1   + (size_t)M * HID;           // [B,N,HID]    = 49152

    const dim3 gridProj(NH * HID / 16, M / 16);     // (64, 24)
    const dim3 gridAttn(N, NH, B);                  // (192, 8, 2)

    // ---- GAT layer 1 (mask active, W1/a1), then ELU ----
    wmma_gemm_f32<0><<<gridProj, 32, 0, stream>>>(x, W1, nullptr, g, M, NH * HID, IN_F);
    gat_attention<<<gridAttn, 128, 0, stream>>>(g, adj, mask, a1, agg, 1);
    head_mean_act<<<(M * HID) / 256, 256, 0, stream>>>(agg, h, 1);

    // ---- GAT layer 2 (W2/a2), then ELU ----
    wmma_gemm_f32<0><<<gridProj, 32, 0, stream>>>(h, W2, nullptr, g, M, NH * HID, HID);
    gat_attention<<<gridAttn, 128, 0, stream>>>(g, adj, nullptr, a2, agg, 0);
    head_mean_act<<<(M * HID) / 256, 256, 0, stream>>>(agg, h, 1);

    // ---- GAT layer 3 (shared W2/a2), no ELU ----
    wmma_gemm_f32<0><<<gridProj, 32, 0, stream>>>(h, W2, nullptr, g, M, NH * HID, HID);
    gat_attention<<<gridAttn, 128, 0, stream>>>(g, adj, nullptr, a2, agg, 0);
    head_mean_act<<<(M * HID) / 256, 256, 0, stream>>>(agg, h, 0);

    // ---- residual concat ----
    concat_residual<<<(M * (HID + 4)) / 256, 256, 0, stream>>>(h, x, hcat);

    // ---- MLP head ----
    wmma_gemm_f32<1><<<dim3(HID / 16, M / 16), 32, 0, stream>>>(hcat, Wl1, bl1, m1, M, HID, HID + 4);
    wmma_gemm_f32<1><<<dim3(HID / 16, M / 16), 32, 0, stream>>>(m1,   Wl2, bl2, m2, M, HID, HID);
    wmma_gemm_f32<0><<<dim3(NC  / 16, M / 16), 32, 0, stream>>>(m2,   Wl3, bl3, out, M, NC, HID);

    (void)in_sizes; (void)n_in; (void)out_size; (void)ws_size;
}